// Co_attention_layer_4312147165639
// MI455X (gfx1250) — compile-verified
//
#include <hip/hip_runtime.h>
#include <math.h>

typedef __attribute__((ext_vector_type(2))) float v2f;
typedef __attribute__((ext_vector_type(8))) float v8f;

#define NUM_FEA 32
#define EMB_DIM 8
#define OUT_DIM 16
#define WAVES_PER_BLOCK 8

// ds_swizzle immediates (group-of-32): new_lane = ((lane & and) | or) ^ xor
//   imm = xor[14:10] | or[9:5] | and[4:0]
#define SWZ(v, imm) __int_as_float(__builtin_amdgcn_ds_swizzle(__float_as_int(v), (imm)))
#define SWZ_X1  0x041F   // xor 1,  and 0x1f
#define SWZ_X2  0x081F   // xor 2
#define SWZ_X4  0x101F   // xor 4
#define SWZ_X8  0x201F   // xor 8
#define SWZ_X16 0x401F   // xor 16 (swap halves)
#define SWZ_BCL 0x000F   // src = lane & 15          -> broadcast left  feature
#define SWZ_BCR 0x020F   // src = 16 | (lane & 15)   -> broadcast right feature

__device__ __forceinline__ float fast_tanh(float v) {
#if __has_builtin(__builtin_amdgcn_tanhf)
    return __builtin_amdgcn_tanhf(v);   // v_tanh_f32 on gfx1250
#else
    return tanhf(v);
#endif
}

// sum of 16 distinct values replicated across both 16-lane halves
__device__ __forceinline__ float sum16(float v) {
    v += SWZ(v, SWZ_X1);
    v += SWZ(v, SWZ_X2);
    v += SWZ(v, SWZ_X4);
    v += SWZ(v, SWZ_X8);
    return v;
}

// One wave32 per row. G(16x16) = (left @ M) @ right^T via two
// V_WMMA_F32_16X16X4_F32 (K = 8 split into 2x4), fp32 end to end.
__global__ __launch_bounds__(256)
void coattn_kernel(const float* __restrict__ x,
                   const float* __restrict__ emb_W,
                   const float* __restrict__ emb_b,
                   const float* __restrict__ M,
                   const float* __restrict__ dense_W,
                   const float* __restrict__ dense_b,
                   float* __restrict__ out,
                   int nrows) {
    __shared__ float sWM[128];    // emb_W[:16] @ M   (16 x 8)
    __shared__ float sbM[128];    // emb_b[:16] @ M   (16 x 8)
    __shared__ float sWr[128];    // emb_W[16:]       (16 x 8)
    __shared__ float sbr[128];    // emb_b[16:]       (16 x 8)
    __shared__ float sdW[512];    // dense_W          (32 x 16)
    __shared__ float sdb[16];     // dense_b
    __shared__ float sT[WAVES_PER_BLOCK][16 * 20];  // G^T, padded rows (20 words)
    __shared__ float out_s[WAVES_PER_BLOCK][32];    // gated row x*a

    const int tid = threadIdx.x;

    // ---- prologue: fold M into the left-half embedding, stage weights ----
    if (tid < 128) {
        const int l = tid >> 3, f = tid & 7;
        float aw = 0.f, ab = 0.f;
        #pragma unroll
        for (int e = 0; e < EMB_DIM; ++e) {
            const float mv = M[e * EMB_DIM + f];
            aw = fmaf(emb_W[l * EMB_DIM + e], mv, aw);
            ab = fmaf(emb_b[l * EMB_DIM + e], mv, ab);
        }
        sWM[tid] = aw;
        sbM[tid] = ab;
        sWr[tid] = emb_W[128 + tid];
        sbr[tid] = emb_b[128 + tid];
    }
    sdW[tid]       = dense_W[tid];
    sdW[tid + 256] = dense_W[tid + 256];
    if (tid < OUT_DIM) sdb[tid] = dense_b[tid];
    __syncthreads();

    const int lane = tid & 31;
    const int wave = tid >> 5;
    int b = blockIdx.x * WAVES_PER_BLOCK + wave;
    const bool valid = (b < nrows);
    if (!valid) b = nrows - 1;            // clamp: keep EXEC all-1s for WMMA

    const int m16  = lane & 15;           // A row (M) / B col (N) / C col (N)
    const int half = lane >> 4;           // K sub-slot within a WMMA operand

    // Row of x: one scalar per lane, 128B coalesced per wave
    const float xv = x[(size_t)b * NUM_FEA + lane];
    const float xl = SWZ(xv, SWZ_BCL);    // x[b, m16]      (left features)
    const float xr = SWZ(xv, SWZ_BCR);    // x[b, 16 + m16] (right features)

    // ---- G_pre = left' (16x8) @ right^T (8x16) : two 16x16x4 f32 WMMAs ----
    // A layout (16x4 f32): lanes 0-15 M=0..15; VGPR r, half h -> K = 4w+2h+r
    v8f c = {};
    #pragma unroll
    for (int w = 0; w < 2; ++w) {
        const int k0 = w * 4 + half * 2;
        v2f a, bmat;
        a.x    = fmaf(xl, sWM[m16 * 8 + k0],     sbM[m16 * 8 + k0]);
        a.y    = fmaf(xl, sWM[m16 * 8 + k0 + 1], sbM[m16 * 8 + k0 + 1]);
        bmat.x = fmaf(xr, sWr[m16 * 8 + k0],     sbr[m16 * 8 + k0]);
        bmat.y = fmaf(xr, sWr[m16 * 8 + k0 + 1], sbr[m16 * 8 + k0 + 1]);
        c = __builtin_amdgcn_wmma_f32_16x16x4_f32(
                /*neg_a=*/false, a, /*neg_b=*/false, bmat,
                /*c_mod=*/(short)0, c, /*reuse_a=*/false, /*reuse_b=*/false);
    }

    // G = tanh(G_pre); C layout: lane L(<16) VGPR v -> (M=v,   N=L)
    //                  lane L+16 VGPR v -> (M=v+8, N=L)
    float g[8];
    #pragma unroll
    for (int v = 0; v < 8; ++v) g[v] = fast_tanh(c[v]);

    // ---- lr path: colsum over M (G.sum(axis=1)), indexed by r = N = m16 ----
    float cs = 0.f;
    #pragma unroll
    for (int v = 0; v < 8; ++v) cs += g[v];
    cs += SWZ(cs, SWZ_X16);               // add partner half's 8 M-rows
    // softmax over 16 r-values; inputs bounded in [-16,16] -> no max needed
    const float e_lr = __expf(cs);
    const float lr_lane = e_lr / sum16(e_lr);   // lr[r = m16]

    // ---- transpose G into LDS so rowsum has the same shape as colsum ----
    // sT[N][M] = G[M][N]; lane writes 8 contiguous words (two b128 stores)
    {
        float* dst = &sT[wave][m16 * 20 + half * 8];
        *(float4*)(dst)     = make_float4(g[0], g[1], g[2], g[3]);
        *(float4*)(dst + 4) = make_float4(g[4], g[5], g[6], g[7]);
    }
    __syncthreads();

    // tg[v] = G[M=m16][N=half*8+v]
    float rs = 0.f;
    #pragma unroll
    for (int v = 0; v < 8; ++v) rs += sT[wave][(half * 8 + v) * 20 + m16];
    rs += SWZ(rs, SWZ_X16);               // full rowsum[l = m16], dup halves
    const float e_rl = __expf(rs);
    const float rl_lane = e_rl / sum16(e_rl);   // rl[l = m16]

    // ---- gate: a = [lr | rl]; lane 16+l already holds rl[l] ----
    const float aval = (lane < 16) ? lr_lane : rl_lane;
    out_s[wave][lane] = xv * aval;
    __syncthreads();

    // ---- dense: y[n] = relu(sum_k out[k]*dW[k][n] + db[n]) ----
    const int n = m16;                    // lanes 16-31 duplicate, don't store
    float acc = sdb[n];
    const float4* ov = (const float4*)out_s[wave];
    #pragma unroll
    for (int k4 = 0; k4 < 8; ++k4) {
        const float4 o = ov[k4];
        acc = fmaf(o.x, sdW[(k4 * 4 + 0) * OUT_DIM + n], acc);
        acc = fmaf(o.y, sdW[(k4 * 4 + 1) * OUT_DIM + n], acc);
        acc = fmaf(o.z, sdW[(k4 * 4 + 2) * OUT_DIM + n], acc);
        acc = fmaf(o.w, sdW[(k4 * 4 + 3) * OUT_DIM + n], acc);
    }
    acc = fmaxf(acc, 0.f);

    if (valid && lane < 16)
        out[(size_t)b * OUT_DIM + n] = acc;
}

extern "C" void kernel_launch(void* const* d_in, const int* in_sizes, int n_in,
                              void* d_out, int out_size, void* d_ws, size_t ws_size,
                              hipStream_t stream) {
    const float* x       = (const float*)d_in[0];
    const float* emb_W   = (const float*)d_in[1];
    const float* emb_b   = (const float*)d_in[2];
    const float* M       = (const float*)d_in[3];
    const float* dense_W = (const float*)d_in[4];
    const float* dense_b = (const float*)d_in[5];
    float* out = (float*)d_out;

    const int nrows  = in_sizes[0] / NUM_FEA;                       // 262144
    const int blocks = (nrows + WAVES_PER_BLOCK - 1) / WAVES_PER_BLOCK;
    coattn_kernel<<<blocks, 256, 0, stream>>>(x, emb_W, emb_b, M,
                                              dense_W, dense_b, out, nrows);
}